// SFM_Model_65317862637905
// MI455X (gfx1250) — compile-verified
//
#include <hip/hip_runtime.h>
#include <math.h>

// Problem constants (from reference)
#define Bv 2048
#define Tv 60
#define Dv 6
#define Hv 64
#define Fv 10
// Concatenated GEMM: z (16 x 72) @ Wz (72 x 272) ; 17 N-tiles, 18 K-steps of 4
#define KZ 72
#define NZ 272
#define NTILES 17
#define KSTEPS 18

// ws layout (floats)
#define WS_PACKEDB 0
#define WS_BIAS   (NTILES * KSTEPS * 64)     // 19584
#define WS_TRIG   (WS_BIAS + NZ)             // 19856 (even -> 8B aligned)

typedef __attribute__((ext_vector_type(2))) float v2f;
typedef __attribute__((ext_vector_type(8))) float v8f;

__device__ __forceinline__ float hsig(float v) {
    return fminf(fmaxf(v * 0.16666667f + 0.5f, 0.0f), 1.0f);
}

// Element of the virtual 72x272 concatenated weight matrix.
// cols: [0,64)=i  [64,128)=ste  [128,192)=c  [192,256)=o  [256,266)=fre, rest 0
// rows: [0,64)=U rows, [64,70)=W rows, [70,72)=0
__device__ float wz_elem(int k, int n,
                         const float* Ui, const float* Uste, const float* Uc,
                         const float* Uo, const float* Ufre,
                         const float* Wi, const float* Wste, const float* Wc,
                         const float* Wo, const float* Wfre) {
    const float *U, *W;
    int col, nc;
    if      (n < 64)  { U = Ui;   W = Wi;   col = n;       nc = Hv; }
    else if (n < 128) { U = Uste; W = Wste; col = n - 64;  nc = Hv; }
    else if (n < 192) { U = Uc;   W = Wc;   col = n - 128; nc = Hv; }
    else if (n < 256) { U = Uo;   W = Wo;   col = n - 192; nc = Hv; }
    else {
        int m = n - 256;
        if (m >= Fv) return 0.0f;
        U = Ufre; W = Wfre; col = m; nc = Fv;
    }
    if (k < Hv)      return U[k * nc + col];
    if (k < Hv + Dv) return W[(k - Hv) * nc + col];
    return 0.0f;
}

// Pre-pack Wz into B-fragment order for V_WMMA_F32_16X16X4_F32:
// lane L of fragment (nt,ks) holds Wz[k0][n], Wz[k0+1][n] with
// k0 = ks*4 + 2*(L/16), n = nt*16 + L%16. One b64 load per WMMA in main loop.
__global__ void pack_frags(const float* Ui, const float* Uste, const float* Uc,
                           const float* Uo, const float* Ufre,
                           const float* Wi, const float* Wste, const float* Wc,
                           const float* Wo, const float* Wfre,
                           float* packed) {
    int idx  = blockIdx.x;                 // nt*KSTEPS + ks  (0..305)
    int nt   = idx / KSTEPS;
    int ks   = idx % KSTEPS;
    int lane = threadIdx.x;
    int n    = nt * 16 + (lane & 15);
    int k0   = ks * 4 + ((lane >> 4) << 1);
    float* o = packed + ((size_t)idx * 32 + lane) * 2;
    o[0] = wz_elem(k0,     n, Ui, Uste, Uc, Uo, Ufre, Wi, Wste, Wc, Wo, Wfre);
    o[1] = wz_elem(k0 + 1, n, Ui, Uste, Uc, Uo, Ufre, Wi, Wste, Wc, Wo, Wfre);
}

// biascat + trig table (cos/sin of 2*pi*(t+1)*f/F for all t,f) in one pass.
__global__ void pack_small(const float* bi, const float* bste, const float* bc,
                           const float* bo, const float* bfre, float* ws) {
    int n = threadIdx.x;                   // 0..871 used: 272 bias + 600 trig
    if (n < NZ) {
        float v;
        if      (n < 64)  v = bi[n];
        else if (n < 128) v = bste[n - 64];
        else if (n < 192) v = bc[n - 128];
        else if (n < 256) v = bo[n - 192];
        else { int m = n - 256; v = (m < Fv) ? bfre[m] : 0.0f; }
        ws[WS_BIAS + n] = v;
    }
    int q = n - NZ;                        // 0..599 -> (t,f)
    if (q >= 0 && q < Tv * Fv) {
        int t = q / Fv, f = q % Fv;
        float ang = 6.283185307179586f * (float)(t + 1) * (float)f * (1.0f / (float)Fv);
        ws[WS_TRIG + q * 2]     = cosf(ang);
        ws[WS_TRIG + q * 2 + 1] = sinf(ang);
    }
}

__global__ __launch_bounds__(128)
void sfm_main(const float* __restrict__ x, const float* __restrict__ ws,
              const float* __restrict__ Ua, const float* __restrict__ ba,
              const float* __restrict__ Wp, const float* __restrict__ bp,
              float* __restrict__ out) {
    __shared__ float ls_z[16][74];    // [h(64) | x(6) | pad0(2)] rows, stride 74 for bank spread
    __shared__ float ls_g[16][276];   // gate pre-activations (272 used)
    __shared__ float ls_wp[Hv];

    const int tid  = threadIdx.x;     // 0..127 (4 waves)
    const int lane = tid & 31;
    const int wave = tid >> 5;
    const int r0   = blockIdx.x * 16; // batch-row tile

    const float* packedB = ws + WS_PACKEDB;
    const float* biascat = ws + WS_BIAS;
    const v2f*   trig    = (const v2f*)(ws + WS_TRIG);

    if (tid < Hv) ls_wp[tid] = Wp[tid];
    for (int idx = tid; idx < 16 * 74; idx += 128) (&ls_z[0][0])[idx] = 0.0f;

    // Per-lane constants
    float ua_r[Fv];
    #pragma unroll
    for (int f = 0; f < Fv; ++f) ua_r[f] = Ua[f];
    float ba_r[8];
    #pragma unroll
    for (int i = 0; i < 8; ++i) { int p = i * 128 + tid; ba_r[i] = ba[p & 63]; }

    // Spectral state: lane owns 8 (row,h) pairs x 10 freqs, kept in VGPRs.
    float Sre[8][Fv], Sim[8][Fv];
    #pragma unroll
    for (int i = 0; i < 8; ++i)
        #pragma unroll
        for (int f = 0; f < Fv; ++f) { Sre[i][f] = 0.0f; Sim[i][f] = 0.0f; }

    __syncthreads();

    for (int t = 0; t < Tv; ++t) {
        // ---- stage x_t into z columns 64..69 ----
        if (tid < 16 * Dv) {
            int r = tid / Dv, d = tid % Dv;
            ls_z[r][Hv + d] = x[((size_t)(r0 + r) * Tv + t) * Dv + d];
        }
        __syncthreads();

        // ---- GEMM phase: z(16x72) @ Wz(72x272) via fp32 WMMA 16x16x4 ----
        // A fragments (shared across this wave's N-tiles), ISA 16x4 f32 layout.
        v2f afrag[KSTEPS];
        {
            int arow = lane & 15;
            int asub = (lane >> 4) << 1;
            #pragma unroll
            for (int ks = 0; ks < KSTEPS; ++ks)
                afrag[ks] = *(const v2f*)&ls_z[arow][ks * 4 + asub];
        }
        #pragma unroll
        for (int j = 0; j < 5; ++j) {
            int nt = wave + j * 4;          // wave-uniform => EXEC stays all-ones
            if (nt < NTILES) {
                float bias = biascat[nt * 16 + (lane & 15)];
                // Two accumulators -> two independent 9-deep WMMA chains (ILP).
                v8f acc0, acc1;
                #pragma unroll
                for (int v = 0; v < 8; ++v) { acc0[v] = bias; acc1[v] = 0.0f; }
                const v2f* bptr = (const v2f*)packedB + (size_t)nt * KSTEPS * 32 + lane;
                #pragma unroll
                for (int ks = 0; ks < KSTEPS; ks += 2) {
                    v2f bf0 = bptr[(size_t)ks * 32];
                    v2f bf1 = bptr[(size_t)(ks + 1) * 32];
                    acc0 = __builtin_amdgcn_wmma_f32_16x16x4_f32(
                        false, afrag[ks], false, bf0, (short)0, acc0, false, false);
                    acc1 = __builtin_amdgcn_wmma_f32_16x16x4_f32(
                        false, afrag[ks + 1], false, bf1, (short)0, acc1, false, false);
                }
                // C layout: VGPR v -> row v + 8*(lane/16), col lane%16
                int ncol = nt * 16 + (lane & 15);
                int rb   = (lane >> 4) * 8;
                #pragma unroll
                for (int v = 0; v < 8; ++v) ls_g[rb + v][ncol] = acc0[v] + acc1[v];
            }
        }
        __syncthreads();

        // ---- element-wise SFM state update (state in registers) ----
        float cosv[Fv], sinv[Fv];
        #pragma unroll
        for (int f = 0; f < Fv; ++f) {
            v2f cs = trig[t * Fv + f];      // uniform, L2-hot
            cosv[f] = cs.x;
            sinv[f] = cs.y;
        }
        #pragma unroll
        for (int i = 0; i < 8; ++i) {
            int p = i * 128 + tid;           // 0..1023 -> (r, h)
            int r = p >> 6, h = p & 63;
            float gi  = ls_g[r][h];
            float gst = ls_g[r][64 + h];
            float gc  = ls_g[r][128 + h];
            float go  = ls_g[r][192 + h];
            float iv  = hsig(gi);
            float stv = hsig(gst);
            float ov  = hsig(go);
            float cv  = iv * tanhf(gc);
            float Adot = 0.0f;
            #pragma unroll
            for (int f = 0; f < Fv; ++f) {
                float frev = hsig(ls_g[r][256 + f]);
                float fv   = stv * frev;
                float sre  = fmaf(fv, Sre[i][f], cv * cosv[f]);
                float sim  = fmaf(fv, Sim[i][f], cv * sinv[f]);
                Sre[i][f] = sre;
                Sim[i][f] = sim;
                Adot = fmaf(fmaf(sre, sre, sim * sim), ua_r[f], Adot);
            }
            float a = tanhf(Adot + ba_r[i]);
            ls_z[r][h] = ov * a;             // new h, becomes next step's A input
        }
        // no barrier here: next iteration's x-stage barrier orders h-writes
        // before A-fragment reads, and gate reads before next gate writes.
    }
    __syncthreads();

    // ---- projection: out[r] = h[r,:] . W_p + b_p ----
    if (tid < 16) {
        float acc = bp[0];
        #pragma unroll 8
        for (int h = 0; h < Hv; ++h) acc = fmaf(ls_z[tid][h], ls_wp[h], acc);
        out[r0 + tid] = acc;
    }
}

extern "C" void kernel_launch(void* const* d_in, const int* in_sizes, int n_in,
                              void* d_out, int out_size, void* d_ws, size_t ws_size,
                              hipStream_t stream) {
    (void)in_sizes; (void)n_in; (void)out_size; (void)ws_size;
    const float* x     = (const float*)d_in[0];
    const float* W_i   = (const float*)d_in[1];
    const float* U_i   = (const float*)d_in[2];
    const float* b_i   = (const float*)d_in[3];
    const float* W_ste = (const float*)d_in[4];
    const float* U_ste = (const float*)d_in[5];
    const float* b_ste = (const float*)d_in[6];
    const float* W_fre = (const float*)d_in[7];
    const float* U_fre = (const float*)d_in[8];
    const float* b_fre = (const float*)d_in[9];
    const float* W_c   = (const float*)d_in[10];
    const float* U_c   = (const float*)d_in[11];
    const float* b_c   = (const float*)d_in[12];
    const float* W_o   = (const float*)d_in[13];
    const float* U_o   = (const float*)d_in[14];
    const float* b_o   = (const float*)d_in[15];
    const float* U_a   = (const float*)d_in[16];
    const float* b_a   = (const float*)d_in[17];
    const float* W_p   = (const float*)d_in[18];
    const float* b_p   = (const float*)d_in[19];

    float* ws = (float*)d_ws;

    pack_frags<<<NTILES * KSTEPS, 32, 0, stream>>>(
        U_i, U_ste, U_c, U_o, U_fre, W_i, W_ste, W_c, W_o, W_fre, ws + WS_PACKEDB);
    pack_small<<<1, 1024, 0, stream>>>(b_i, b_ste, b_c, b_o, b_fre, ws);
    sfm_main<<<Bv / 16, 128, 0, stream>>>(
        x, ws, U_a, b_a, W_p, b_p, (float*)d_out);
}